// PagedAttention_16252156248233
// MI455X (gfx1250) — compile-verified
//
#include <hip/hip_runtime.h>

typedef __attribute__((ext_vector_type(2))) float v2f;
typedef __attribute__((ext_vector_type(8))) float v8f;

#define ATTN_SCALE 0.08838834764831845f

__device__ __forceinline__ v8f wmma_f32(v2f a, v2f b, v8f c) {
    // D(16x16) = A(16x4) * B(4x16) + C
    return __builtin_amdgcn_wmma_f32_16x16x4_f32(
        /*neg_a=*/false, a, /*neg_b=*/false, b,
        /*c_mod=*/(short)0, c, /*reuse_a=*/false, /*reuse_b=*/false);
}

__launch_bounds__(256)
__global__ void paged_attn_f32_wmma(const float* __restrict__ query,
                                    const float* __restrict__ key_cache,
                                    const float* __restrict__ value_cache,
                                    const int*  __restrict__ block_tables,
                                    const int*  __restrict__ context_lens,
                                    float* __restrict__ out) {
    constexpr int KVH = 8, G = 4, D = 128, BLK = 16, BMAX = 128, H = 32;
    const int s    = blockIdx.x >> 3;   // sequence
    const int kvh  = blockIdx.x & 7;    // kv head
    const int tid  = threadIdx.x;
    const int wv   = tid >> 5;          // 8 waves per workgroup
    const int lane = tid & 31;
    const int l16  = lane & 15;         // column index g (and token-in-row / d-row)
    const int hi   = lane >> 4;         // half-wave select
    const bool lowhalf = (hi == 0);

    __shared__ float lds_acc[D * 16];   // out^T accumulator [d][g] (8 KB)
    __shared__ float lds_m[8 * 16];
    __shared__ float lds_l[8 * 16];
    __shared__ float lds_rl[16];

    for (int i = tid; i < D * 16; i += 256) lds_acc[i] = 0.0f;

    const int ctx    = context_lens[s];
    const int ntiles = (ctx + BLK - 1) / BLK;

    // Preload Q^T in WMMA-B layout: qreg[dc] lane L: v0=Q[g=l16][dc*4+2*hi], v1=next dim.
    // Columns g>=4 zeroed (pad of the GQA group to the 16-wide tile).
    const float* qbase = query + ((size_t)s * H + (size_t)kvh * G) * D;
    const bool gok = (l16 < G);
    v2f qreg[32];
    #pragma unroll
    for (int dc = 0; dc < 32; ++dc) {
        const int d0 = dc * 4 + 2 * hi;
        v2f qv; qv.x = 0.0f; qv.y = 0.0f;
        if (gok) {
            qv.x = qbase[l16 * D + d0];
            qv.y = qbase[l16 * D + d0 + 1];
        }
        qreg[dc] = qv;
    }

    v8f zero8 = {};
    v8f acc_o[8];                       // out^T tiles: row d = r + 8*hi (+16*dt), col g = l16
    #pragma unroll
    for (int dt = 0; dt < 8; ++dt) acc_o[dt] = zero8;
    float m = -3.0e38f;                 // running max over tokens, per column g (per-lane scalar)
    float l = 0.0f;                     // running sum, per column g

    const int* bt = block_tables + (size_t)s * BMAX;

    for (int tile = wv; tile < ntiles; tile += 8) {
        const int bid = bt[tile];

        // ---- S^T(16 tok x 16 g) = K(16x128) * Q^T(128x16), 32 WMMAs of K-depth 4 ----
        const float* kp = key_cache + ((size_t)bid * KVH + kvh) * (D * BLK);
        v8f sa = zero8;
        #pragma unroll
        for (int dc = 0; dc < 32; ++dc) {
            const int d0 = dc * 4 + 2 * hi;
            v2f kv;
            kv.x = kp[d0 * BLK + l16];          // A: row=token l16, K-dim d0
            kv.y = kp[(d0 + 1) * BLK + l16];    //    K-dim d0+1 (64B apart, coalesced per half)
            sa = wmma_f32(kv, qreg[dc], sa);
        }

        // ---- online softmax over token rows (per-lane column g) ----
        float sv[8];
        float smax = -3.0e38f;
        #pragma unroll
        for (int r = 0; r < 8; ++r) {
            const int tok = tile * BLK + r + 8 * hi;
            const float x = (tok < ctx) ? sa[r] * ATTN_SCALE : -3.0e38f;
            sv[r] = x;
            smax = fmaxf(smax, x);
        }
        smax = fmaxf(smax, __shfl_xor(smax, 16, 32));
        const float m_new = fmaxf(m, smax);
        const float alpha = __expf(m - m_new);
        float p8[8];
        float lsum = 0.0f;
        #pragma unroll
        for (int r = 0; r < 8; ++r) {
            const int tok = tile * BLK + r + 8 * hi;
            const float pv = (tok < ctx) ? __expf(sv[r] - m_new) : 0.0f;
            p8[r] = pv;
            lsum += pv;
        }
        lsum += __shfl_xor(lsum, 16, 32);
        l = l * alpha + lsum;
        m = m_new;
        #pragma unroll
        for (int dt = 0; dt < 8; ++dt) acc_o[dt] = acc_o[dt] * alpha;

        // ---- repack P^T (C/D layout) into WMMA-B chunks (4 tokens x 16 g) ----
        v2f Bp[4];
        #pragma unroll
        for (int tc = 0; tc < 2; ++tc) {        // tokens 0..7 live in half 0
            const float a0 = p8[tc * 4 + 0], a1 = p8[tc * 4 + 1];
            const float a2 = p8[tc * 4 + 2], a3 = p8[tc * 4 + 3];
            const float s2 = __shfl_xor(a2, 16, 32);
            const float s3 = __shfl_xor(a3, 16, 32);
            v2f b; b.x = lowhalf ? a0 : s2; b.y = lowhalf ? a1 : s3;
            Bp[tc] = b;
        }
        #pragma unroll
        for (int tc = 2; tc < 4; ++tc) {        // tokens 8..15 live in half 1
            const int rr = (tc - 2) * 4;
            const float a0 = p8[rr + 0], a1 = p8[rr + 1];
            const float a2 = p8[rr + 2], a3 = p8[rr + 3];
            const float s0 = __shfl_xor(a0, 16, 32);
            const float s1 = __shfl_xor(a1, 16, 32);
            v2f b; b.x = lowhalf ? s0 : a2; b.y = lowhalf ? s1 : a3;
            Bp[tc] = b;
        }

        // ---- out^T(128 d x 16 g) += V^T(128x16) * P^T(16x16), 32 WMMAs ----
        const float* vp = value_cache + ((size_t)bid * KVH + kvh) * (D * BLK);
        #pragma unroll
        for (int dt = 0; dt < 8; ++dt) {
            v8f o = acc_o[dt];
            #pragma unroll
            for (int tc = 0; tc < 4; ++tc) {
                const float* va = vp + (dt * 16 + l16) * BLK + tc * 4 + 2 * hi;
                v2f av; av.x = va[0]; av.y = va[1];   // two consecutive tokens -> b64 load
                o = wmma_f32(av, Bp[tc], o);
            }
            acc_o[dt] = o;
        }
    }

    // ---- cross-wave flash merge via LDS ----
    if (hi == 0) {
        lds_m[wv * 16 + l16] = m;
        lds_l[wv * 16 + l16] = l;
    }
    __syncthreads();
    float M = -3.0e38f;
    #pragma unroll
    for (int w = 0; w < 8; ++w) M = fmaxf(M, lds_m[w * 16 + l16]);
    float Lt = 0.0f;
    #pragma unroll
    for (int w = 0; w < 8; ++w) Lt += __expf(lds_m[w * 16 + l16] - M) * lds_l[w * 16 + l16];
    const float myalpha = __expf(m - M);
    if (tid < 16) lds_rl[tid] = 1.0f / Lt;   // wave 0, lane == g

    #pragma unroll
    for (int dt = 0; dt < 8; ++dt) {
        #pragma unroll
        for (int r = 0; r < 8; ++r) {
            const int d = dt * 16 + r + 8 * hi;
            atomicAdd(&lds_acc[d * 16 + l16], acc_o[dt][r] * myalpha);
        }
    }
    __syncthreads();

    // ---- normalized, coalesced store: out[s][kvh*4+g][d] ----
    float* ob = out + ((size_t)s * H + (size_t)kvh * G) * D;
    for (int i = tid; i < G * D; i += 256) {
        const int g = i >> 7;       // 0..3
        const int d = i & (D - 1);  // 0..127
        ob[g * D + d] = lds_acc[d * 16 + g] * lds_rl[g];
    }
}

extern "C" void kernel_launch(void* const* d_in, const int* in_sizes, int n_in,
                              void* d_out, int out_size, void* d_ws, size_t ws_size,
                              hipStream_t stream) {
    const float* query       = (const float*)d_in[0];
    const float* key_cache   = (const float*)d_in[1];
    const float* value_cache = (const float*)d_in[2];
    const int*   block_tables = (const int*)d_in[3];
    const int*   context_lens = (const int*)d_in[4];
    float* out = (float*)d_out;

    const int num_seqs = in_sizes[4];          // 32
    dim3 grid(num_seqs * 8);                   // (seq, kv_head)
    dim3 block(256);                           // 8 wave32
    hipLaunchKernelGGL(paged_attn_f32_wmma, grid, block, 0, stream,
                       query, key_cache, value_cache, block_tables, context_lens, out);
}